// DRNN_65429531787924
// MI455X (gfx1250) — compile-verified
//
#include <hip/hip_runtime.h>
#include <hip/hip_bf16.h>
#include <math.h>

// Problem constants (match reference)
#define B_   128
#define T_   40
#define E_   512
#define H_   512
#define V_   10000
#define FC_  2048
#define RS   516   // LDS row stride (floats): 516%64==4 -> conflict-free 16-row A-tile reads

typedef float v2f __attribute__((ext_vector_type(2)));
typedef float v8f __attribute__((ext_vector_type(8)));

__device__ __forceinline__ v8f zero8() {
  v8f z = {0.f,0.f,0.f,0.f,0.f,0.f,0.f,0.f};
  return z;
}

// D(16x16,f32) = A(16x4,f32) * B(4x16,f32) + C   (full fp32 WMMA)
__device__ __forceinline__ v8f wmma4(v2f a, v2f b, v8f c) {
  return __builtin_amdgcn_wmma_f32_16x16x4_f32(false, a, false, b, (short)0, c, false, false);
}

__device__ __forceinline__ v2f ldg2(const float* p) { return *(const v2f*)p; }
__device__ __forceinline__ float sigf(float x) { return 1.0f / (1.0f + expf(-x)); }

// ---------------------------------------------------------------------------
// Persistent recurrence kernel: 8 blocks x 512 threads (16 waves), 16 batch
// rows per block, runs all 40 timesteps. Gate GEMM: wave w owns columns
// q*512 + w*32 + {0..31} of each gate quadrant q (i,f,g,o) so the LSTM
// pointwise math runs entirely on WMMA accumulator fragments.
// ---------------------------------------------------------------------------
__global__ __launch_bounds__(512) void drnn_recurrent(
    const int* __restrict__ word_idx, const int* __restrict__ father_idx,
    const float* __restrict__ fc_feats, const float* __restrict__ embed,
    const float* __restrict__ fc_w, const float* __restrict__ fc_b,
    const float* __restrict__ a_wih, const float* __restrict__ a_whh,
    const float* __restrict__ a_bih, const float* __restrict__ a_bhh,
    const float* __restrict__ f_wih, const float* __restrict__ f_whh,
    const float* __restrict__ f_bih, const float* __restrict__ f_bhh,
    const float* __restrict__ pred_w, const float* __restrict__ pred_b,
    float* __restrict__ hS, float* __restrict__ cS,
    float* __restrict__ OUT, float* __restrict__ XA0)
{
  __shared__ float sHA[16 * RS];   // new ancestral h tile (pred input)
  __shared__ float sHF[16 * RS];   // fraternal h tile (persists across steps)
  __shared__ int   sFC[16];        // father column per row
  __shared__ int   sWA[16];        // parent word id per row
  __shared__ int   sWP[16];        // previous word id per row

  const int tid = threadIdx.x;
  const int w   = tid >> 5;        // wave 0..15
  const int l   = tid & 31;        // lane
  const int ln  = l & 15;          // A-row / B-col / D-col lane index
  const int hi  = l >> 4;          // half-wave (K-pair / D-row-half select)
  const int kh  = hi << 1;         // 0 or 2
  const int b0  = blockIdx.x << 4; // first batch row of this block

  // Hoisted bias fragments (depend only on column -> on lane, not on t)
  float bA[8], bF[8];
  #pragma unroll
  for (int q = 0; q < 4; ++q) {
    #pragma unroll
    for (int tt = 0; tt < 2; ++tt) {
      int n = q * 512 + w * 32 + tt * 16 + ln;
      bA[q * 2 + tt] = a_bih[n] + a_bhh[n];
      bF[q * 2 + tt] = f_bih[n] + f_bhh[n];
    }
  }
  float bP[2];
  bP[0] = pred_b[w * 32 + ln];
  bP[1] = pred_b[w * 32 + 16 + ln];

  // ---- Stage 0: XA0 = fc_feats @ fc_w.T + fc_b (this block's 16 rows) ----
  {
    v8f a0 = zero8(), a1 = zero8();
    const float* arow = fc_feats + (size_t)(b0 + ln) * FC_;
    const int n0 = w * 32, n1 = n0 + 16;
    for (int k0 = 0; k0 < FC_; k0 += 4) {
      v2f a = ldg2(arow + k0 + kh);
      a0 = wmma4(a, ldg2(fc_w + (size_t)(n0 + ln) * FC_ + k0 + kh), a0);
      a1 = wmma4(a, ldg2(fc_w + (size_t)(n1 + ln) * FC_ + k0 + kh), a1);
    }
    const float fb0 = fc_b[n0 + ln], fb1 = fc_b[n1 + ln];
    #pragma unroll
    for (int r = 0; r < 8; ++r) {
      const int m = r + hi * 8;
      XA0[(size_t)(b0 + m) * E_ + n0 + ln] = a0[r] + fb0;
      XA0[(size_t)(b0 + m) * E_ + n1 + ln] = a1[r] + fb1;
    }
  }
  __threadfence();
  __syncthreads();

  // Fraternal cell state lives in registers (this lane owns fixed (m,j) cells)
  float cF[2][8];
  #pragma unroll
  for (int tt = 0; tt < 2; ++tt)
    #pragma unroll
    for (int r = 0; r < 8; ++r) cF[tt][r] = 0.f;

  for (int t = 0; t < T_; ++t) {
    const bool reset = (t == 0) || (((t - 1) % 3) == 0);

    // ---- Stage 1: per-step index gather into LDS ----
    if (tid < 16) {
      if (t > 0) {
        const int b    = b0 + tid;
        const int fcol = father_idx[b * T_ + t];
        sFC[tid] = fcol;
        sWA[tid] = word_idx[b * T_ + fcol];
        sWP[tid] = word_idx[b * T_ + (t - 1)];
      } else {
        sFC[tid] = 0; sWA[tid] = 0; sWP[tid] = 0;
      }
    }
    __syncthreads();

    // ---- Stage 2: ancestral gates = x_a @ a_wih.T + h_par @ a_whh.T ----
    v8f accA[8];
    #pragma unroll
    for (int f = 0; f < 8; ++f) accA[f] = zero8();
    {
      const float* xrow = (t == 0) ? (XA0 + (size_t)(b0 + ln) * E_)
                                   : (embed + (size_t)sWA[ln] * E_);
      for (int k0 = 0; k0 < E_; k0 += 4) {
        v2f a = ldg2(xrow + k0 + kh);
        #pragma unroll
        for (int f = 0; f < 8; ++f) {
          const int n = (f >> 1) * 512 + w * 32 + (f & 1) * 16 + ln;
          accA[f] = wmma4(a, ldg2(a_wih + (size_t)n * E_ + k0 + kh), accA[f]);
        }
      }
      if (t > 0) {
        const float* hrow = hS + ((size_t)(b0 + ln) * T_ + sFC[ln]) * H_;
        for (int k0 = 0; k0 < H_; k0 += 4) {
          v2f a = ldg2(hrow + k0 + kh);
          #pragma unroll
          for (int f = 0; f < 8; ++f) {
            const int n = (f >> 1) * 512 + w * 32 + (f & 1) * 16 + ln;
            accA[f] = wmma4(a, ldg2(a_whh + (size_t)n * H_ + k0 + kh), accA[f]);
          }
        }
      }
    }

    // ---- Stage 3: fraternal gates (zero inputs/state when reset) ----
    v8f accF[8];
    #pragma unroll
    for (int f = 0; f < 8; ++f) accF[f] = zero8();
    if (!reset) {
      const float* xrow = embed + (size_t)sWP[ln] * E_;
      for (int k0 = 0; k0 < E_; k0 += 4) {
        v2f a = ldg2(xrow + k0 + kh);
        #pragma unroll
        for (int f = 0; f < 8; ++f) {
          const int n = (f >> 1) * 512 + w * 32 + (f & 1) * 16 + ln;
          accF[f] = wmma4(a, ldg2(f_wih + (size_t)n * E_ + k0 + kh), accF[f]);
        }
      }
      for (int k0 = 0; k0 < H_; k0 += 4) {
        v2f a = *(const v2f*)&sHF[ln * RS + k0 + kh];
        #pragma unroll
        for (int f = 0; f < 8; ++f) {
          const int n = (f >> 1) * 512 + w * 32 + (f & 1) * 16 + ln;
          accF[f] = wmma4(a, ldg2(f_whh + (size_t)n * H_ + k0 + kh), accF[f]);
        }
      }
    }
    __syncthreads();   // all sHF reads done before overwrite

    // ---- Stage 4: register-resident LSTM pointwise + state stores ----
    #pragma unroll
    for (int tt = 0; tt < 2; ++tt) {
      const int j = w * 32 + tt * 16 + ln;
      #pragma unroll
      for (int r = 0; r < 8; ++r) {
        const int m = r + hi * 8;
        const int b = b0 + m;
        float cprev = 0.f;
        if (t > 0) cprev = cS[((size_t)b * T_ + sFC[m]) * H_ + j];
        const float ig = sigf (accA[0 + tt][r] + bA[0 + tt]);
        const float fg = sigf (accA[2 + tt][r] + bA[2 + tt]);
        const float gg = tanhf(accA[4 + tt][r] + bA[4 + tt]);
        const float og = sigf (accA[6 + tt][r] + bA[6 + tt]);
        const float c2 = fg * cprev + ig * gg;
        const float h2 = og * tanhf(c2);
        sHA[m * RS + j] = h2;
        hS[((size_t)b * T_ + t) * H_ + j] = h2;
        cS[((size_t)b * T_ + t) * H_ + j] = c2;

        const float cpf = reset ? 0.f : cF[tt][r];
        const float i2 = sigf (accF[0 + tt][r] + bF[0 + tt]);
        const float f2 = sigf (accF[2 + tt][r] + bF[2 + tt]);
        const float g2 = tanhf(accF[4 + tt][r] + bF[4 + tt]);
        const float o2 = sigf (accF[6 + tt][r] + bF[6 + tt]);
        const float cf2 = f2 * cpf + i2 * g2;
        cF[tt][r] = cf2;
        sHF[m * RS + j] = o2 * tanhf(cf2);
      }
    }
    __threadfence();   // hS/cS visible to future-step gathers
    __syncthreads();

    // ---- Stage 5: out = tanh(concat(hA,hF) @ pred_w.T + pred_b) ----
    v8f po[2];
    po[0] = zero8(); po[1] = zero8();
    for (int k0 = 0; k0 < H_; k0 += 4) {
      v2f a = *(const v2f*)&sHA[ln * RS + k0 + kh];
      po[0] = wmma4(a, ldg2(pred_w + (size_t)(w * 32      + ln) * 1024 + k0 + kh), po[0]);
      po[1] = wmma4(a, ldg2(pred_w + (size_t)(w * 32 + 16 + ln) * 1024 + k0 + kh), po[1]);
    }
    for (int k0 = 0; k0 < H_; k0 += 4) {
      v2f a = *(const v2f*)&sHF[ln * RS + k0 + kh];
      po[0] = wmma4(a, ldg2(pred_w + (size_t)(w * 32      + ln) * 1024 + 512 + k0 + kh), po[0]);
      po[1] = wmma4(a, ldg2(pred_w + (size_t)(w * 32 + 16 + ln) * 1024 + 512 + k0 + kh), po[1]);
    }
    #pragma unroll
    for (int tt = 0; tt < 2; ++tt) {
      const int n = w * 32 + tt * 16 + ln;
      #pragma unroll
      for (int r = 0; r < 8; ++r) {
        const int m = r + hi * 8;
        OUT[((size_t)(b0 + m) * T_ + t) * H_ + n] = tanhf(po[tt][r] + bP[tt]);
      }
    }
    // barrier-matching analysis: next-step LDS writes are all behind a barrier,
    // so no trailing barrier is required here.
  }
}

// ---------------------------------------------------------------------------
// logits[r, v] = OUT[r, :] @ logit_w[v, :] + logit_b[v]
// grid (320 row-tiles, 10 col groups); each wave: 4 N-tiles reusing each A frag
// ---------------------------------------------------------------------------
__global__ __launch_bounds__(512) void logit_gemm(
    const float* __restrict__ OUT, const float* __restrict__ logit_w,
    const float* __restrict__ logit_b, float* __restrict__ out)
{
  const int tid = threadIdx.x;
  const int w = tid >> 5, l = tid & 31, ln = l & 15, hi = l >> 4, kh = hi << 1;
  const int m0 = blockIdx.x << 4;
  const int tile0 = (blockIdx.y * 16 + w) * 4;   // 625 valid tiles (10000 = 625*16)

  v8f acc[4];
  int n0[4];
  #pragma unroll
  for (int j = 0; j < 4; ++j) {
    acc[j] = zero8();
    const int tl = tile0 + j;
    n0[j] = (tl < 625 ? tl : 624) * 16;          // clamp keeps loads in-bounds
  }
  const float* arow = OUT + (size_t)(m0 + ln) * H_;
  for (int k0 = 0; k0 < H_; k0 += 4) {
    v2f a = ldg2(arow + k0 + kh);
    #pragma unroll
    for (int j = 0; j < 4; ++j)
      acc[j] = wmma4(a, ldg2(logit_w + (size_t)(n0[j] + ln) * H_ + k0 + kh), acc[j]);
  }
  #pragma unroll
  for (int j = 0; j < 4; ++j) {
    if (tile0 + j < 625) {
      const int n = n0[j] + ln;
      const float bb = logit_b[n];
      #pragma unroll
      for (int r = 0; r < 8; ++r) {
        const int m = m0 + r + hi * 8;
        out[(size_t)m * V_ + n] = acc[j][r] + bb;
      }
    }
  }
}

// ---------------------------------------------------------------------------
// In-place log_softmax over the vocab axis; one 256-thread block per row.
// ---------------------------------------------------------------------------
__global__ __launch_bounds__(256) void log_softmax_rows(float* __restrict__ out)
{
  const int r = blockIdx.x;
  float* row = out + (size_t)r * V_;
  const int tid = threadIdx.x;
  __shared__ float red[8];

  float mx = -3.4e38f;
  for (int i = tid; i < V_; i += 256) mx = fmaxf(mx, row[i]);
  #pragma unroll
  for (int off = 16; off; off >>= 1) mx = fmaxf(mx, __shfl_xor(mx, off, 32));
  if ((tid & 31) == 0) red[tid >> 5] = mx;
  __syncthreads();
  float M = red[0];
  #pragma unroll
  for (int j = 1; j < 8; ++j) M = fmaxf(M, red[j]);
  __syncthreads();

  float s = 0.f;
  for (int i = tid; i < V_; i += 256) s += expf(row[i] - M);
  #pragma unroll
  for (int off = 16; off; off >>= 1) s += __shfl_xor(s, off, 32);
  if ((tid & 31) == 0) red[tid >> 5] = s;
  __syncthreads();
  float S = 0.f;
  #pragma unroll
  for (int j = 0; j < 8; ++j) S += red[j];

  const float lse = M + logf(S);
  for (int i = tid; i < V_; i += 256) row[i] = row[i] - lse;
}

// ---------------------------------------------------------------------------
extern "C" void kernel_launch(void* const* d_in, const int* in_sizes, int n_in,
                              void* d_out, int out_size, void* d_ws, size_t ws_size,
                              hipStream_t stream)
{
  const int*   word_idx   = (const int*)  d_in[0];
  const int*   father_idx = (const int*)  d_in[1];
  const float* fc_feats   = (const float*)d_in[2];
  const float* embed      = (const float*)d_in[3];
  const float* fc_w       = (const float*)d_in[4];
  const float* fc_b       = (const float*)d_in[5];
  const float* a_wih      = (const float*)d_in[6];
  const float* a_whh      = (const float*)d_in[7];
  const float* a_bih      = (const float*)d_in[8];
  const float* a_bhh      = (const float*)d_in[9];
  const float* f_wih      = (const float*)d_in[10];
  const float* f_whh      = (const float*)d_in[11];
  const float* f_bih      = (const float*)d_in[12];
  const float* f_bhh      = (const float*)d_in[13];
  const float* pred_w     = (const float*)d_in[14];
  const float* pred_b     = (const float*)d_in[15];
  const float* logit_w    = (const float*)d_in[16];
  const float* logit_b    = (const float*)d_in[17];

  float* ws = (float*)d_ws;
  const size_t HSZ = (size_t)B_ * T_ * H_;
  float* hS  = ws;              // [B,T,H]
  float* cS  = hS + HSZ;        // [B,T,H]
  float* OUT = cS + HSZ;        // [B,T,H]
  float* XA0 = OUT + HSZ;       // [B,H]

  drnn_recurrent<<<dim3(B_ / 16), 512, 0, stream>>>(
      word_idx, father_idx, fc_feats, embed, fc_w, fc_b,
      a_wih, a_whh, a_bih, a_bhh, f_wih, f_whh, f_bih, f_bhh,
      pred_w, pred_b, hS, cS, OUT, XA0);

  logit_gemm<<<dim3((B_ * T_) / 16, 10), 512, 0, stream>>>(
      OUT, logit_w, logit_b, (float*)d_out);

  log_softmax_rows<<<dim3(B_ * T_), 256, 0, stream>>>((float*)d_out);
}